// PeriodicBCFourier_51539607552623
// MI455X (gfx1250) — compile-verified
//
#include <hip/hip_runtime.h>
#include <hip/hip_bf16.h>
#include <stdint.h>

// Reference = transpose -> fft2 -> ifft2 -> real -> transpose-back == identity.
// Optimal kernel: HBM-bandwidth copy (512 MiB total traffic, ~23 us at 23.3 TB/s),
// on the CDNA5 async global<->LDS DMA path (ASYNCcnt-tracked), with non-temporal
// cache hints since the stream (512 MiB) exceeds the 192 MiB L2 (one-touch data).

#define TPB 256        // 8 wave32s per block
#define CHUNKS_PER_T 4 // 4 x 16B per thread per block

__global__ __launch_bounds__(TPB) void PeriodicBCFourier_copy_async_kernel(
    const float* __restrict__ in, float* __restrict__ out)
{
    extern __shared__ char smem[];   // 16 KiB staging: [CHUNKS_PER_T][TPB][16B]
    (void)smem;                      // accessed only via async LDS ops (offset 0 base)

    const unsigned tid = threadIdx.x;
    // 16-byte chunk index for k=0 slot; consecutive lanes -> consecutive 16B chunks.
    const uint64_t chunk0 = (uint64_t)blockIdx.x * (CHUNKS_PER_T * TPB) + tid;

    const uint64_t src = (uint64_t)(uintptr_t)in  + chunk0 * 16u;
    const uint64_t dst = (uint64_t)(uintptr_t)out + chunk0 * 16u;
    const unsigned lds0 = tid * 16u; // LDS byte address of this lane's slot 0

    // Stage 1: 4 async 16B/lane loads, global -> LDS, non-temporal (no L2 retention).
#pragma unroll
    for (int k = 0; k < CHUNKS_PER_T; ++k) {
        const uint64_t gsrc = src + (uint64_t)k * (TPB * 16u);
        const unsigned lds  = lds0 + (unsigned)k * (TPB * 16u);
        asm volatile("global_load_async_to_lds_b128 %0, %1, off th:TH_LOAD_NT"
                     :: "v"(lds), "v"(gsrc)
                     : "memory");
    }

    // Wait for all async loads to land in LDS.
    asm volatile("s_wait_asynccnt 0" ::: "memory");

    // Stage 2: 4 async 16B/lane stores, LDS -> global, non-temporal.
#pragma unroll
    for (int k = 0; k < CHUNKS_PER_T; ++k) {
        const uint64_t gdst = dst + (uint64_t)k * (TPB * 16u);
        const unsigned lds  = lds0 + (unsigned)k * (TPB * 16u);
        asm volatile("global_store_async_from_lds_b128 %0, %1, off th:TH_STORE_NT"
                     :: "v"(gdst), "v"(lds)
                     : "memory");
    }

    // Drain stores before the wave retires (S_ENDPGM also waits, but be explicit).
    asm volatile("s_wait_asynccnt 0" ::: "memory");
}

extern "C" void kernel_launch(void* const* d_in, const int* in_sizes, int n_in,
                              void* d_out, int out_size, void* d_ws, size_t ws_size,
                              hipStream_t stream) {
    (void)in_sizes; (void)n_in; (void)d_ws; (void)ws_size;

    const float* in = (const float*)d_in[0];
    float* out = (float*)d_out;

    // 32*512*512*8 = 67,108,864 floats = 16,777,216 16B chunks.
    // Each block moves CHUNKS_PER_T*TPB = 1024 chunks -> exactly 16384 blocks.
    const uint64_t n_chunks = (uint64_t)out_size / 4u;          // floats -> float4 chunks
    const uint32_t blocks   = (uint32_t)(n_chunks / (CHUNKS_PER_T * TPB));
    const size_t   lds_size = (size_t)CHUNKS_PER_T * TPB * 16u; // 16 KiB

    PeriodicBCFourier_copy_async_kernel<<<dim3(blocks), dim3(TPB), lds_size, stream>>>(in, out);
}